// RNN_Alpha_188978561252
// MI455X (gfx1250) — compile-verified
//
#include <hip/hip_runtime.h>

#define H     2048
#define NIN   128
#define NOUT  32
#define ALPHA 0.2f
// sqrt(2/ALPHA) * SIGMA_REC = sqrt(10) * 0.05
#define NSCALE 0.15811388300841898f

typedef __attribute__((ext_vector_type(2))) float v2f;
typedef __attribute__((ext_vector_type(8))) float v8f;

// ---------------------------------------------------------------------------
// Kernel 1: drive = u @ W_in^T + b_h + NSCALE*noise, written into hidden[]
// WMMA f32 16x16x4. Each wave computes a 16(t) x 64(h) tile (A reused 4x).
// Inner loop is double-buffered so WMMAs never wait on just-issued loads.
// ---------------------------------------------------------------------------
__global__ void drive_kernel(const float* __restrict__ input_tensor,
                             const int*   __restrict__ bsel,
                             const float* __restrict__ noise,
                             const float* __restrict__ W_in,
                             const float* __restrict__ b_h,
                             float*       __restrict__ hidden,
                             int Tlen, int Tfull) {
    const int lane = threadIdx.x & 31;
    const int wave = (blockIdx.x * blockDim.x + threadIdx.x) >> 5;
    const int HG   = H / 64;                 // 32 h-groups of 64 columns
    const int t_tile = wave / HG;
    const int hg     = wave - t_tile * HG;
    if (t_tile * 16 >= Tlen) return;         // wave-uniform: EXEC stays all-1s

    const float* u = input_tensor + (size_t)bsel[0] * (size_t)Tfull * NIN;

    const int m    = lane & 15;
    const int half = lane >> 4;              // lanes 16-31 carry K+2
    int tt = t_tile * 16 + m;
    if (tt >= Tlen) tt = Tlen - 1;           // clamp loads; stores guarded below
    const int n0 = hg * 64 + m;

    const float* arow = u    + (size_t)tt * NIN;
    const float* br0  = W_in + (size_t)(n0     ) * NIN;
    const float* br1  = W_in + (size_t)(n0 + 16) * NIN;
    const float* br2  = W_in + (size_t)(n0 + 32) * NIN;
    const float* br3  = W_in + (size_t)(n0 + 48) * NIN;

    v8f c0 = {}, c1 = {}, c2 = {}, c3 = {};

    // prologue loads for k0 = 0
    int k = 2 * half;
    v2f a  = *(const v2f*)(arow + k);
    v2f b0 = *(const v2f*)(br0  + k);
    v2f b1 = *(const v2f*)(br1  + k);
    v2f b2 = *(const v2f*)(br2  + k);
    v2f b3 = *(const v2f*)(br3  + k);

#pragma unroll 4
    for (int k0 = 0; k0 < NIN; k0 += 4) {
        // issue next iteration's loads before consuming current data
        const int kn = ((k0 + 4) < NIN ? (k0 + 4) : 0) + 2 * half;
        v2f an  = *(const v2f*)(arow + kn);
        v2f b0n = *(const v2f*)(br0  + kn);
        v2f b1n = *(const v2f*)(br1  + kn);
        v2f b2n = *(const v2f*)(br2  + kn);
        v2f b3n = *(const v2f*)(br3  + kn);

        c0 = __builtin_amdgcn_wmma_f32_16x16x4_f32(false, a, false, b0, (short)0, c0, false, false);
        c1 = __builtin_amdgcn_wmma_f32_16x16x4_f32(false, a, false, b1, (short)0, c1, false, false);
        c2 = __builtin_amdgcn_wmma_f32_16x16x4_f32(false, a, false, b2, (short)0, c2, false, false);
        c3 = __builtin_amdgcn_wmma_f32_16x16x4_f32(false, a, false, b3, (short)0, c3, false, false);

        a = an; b0 = b0n; b1 = b1n; b2 = b2n; b3 = b3n;
    }

    const float bh0 = b_h[n0], bh1 = b_h[n0 + 16], bh2 = b_h[n0 + 32], bh3 = b_h[n0 + 48];
#pragma unroll
    for (int i = 0; i < 8; ++i) {
        const int mo = i + 8 * half;         // C/D layout: M = i + 8*(lane>=16)
        const int to = t_tile * 16 + mo;
        if (to < Tlen) {
            const size_t ro = (size_t)to * H;
            hidden[ro + n0     ] = c0[i] + bh0 + NSCALE * noise[ro + n0     ];
            hidden[ro + n0 + 16] = c1[i] + bh1 + NSCALE * noise[ro + n0 + 16];
            hidden[ro + n0 + 32] = c2[i] + bh2 + NSCALE * noise[ro + n0 + 32];
            hidden[ro + n0 + 48] = c3[i] + bh3 + NSCALE * noise[ro + n0 + 48];
        }
    }
}

// ---------------------------------------------------------------------------
// Kernel 2: zero the per-step barrier counters (deterministic each call)
// ---------------------------------------------------------------------------
__global__ void zero_kernel(unsigned* __restrict__ p, int n) {
    int i = blockIdx.x * blockDim.x + threadIdx.x;
    if (i < n) p[i] = 0u;
}

// ---------------------------------------------------------------------------
// Kernel 3: persistent sequential scan. 64 blocks x 256 threads (co-resident).
// Block b owns rows [32b, 32b+32). Per step: stage h_{t-1} in LDS, each wave
// does 4 matvec rows (float4 coalesced from L2-resident W_rec), shuffle-reduce,
// softplus, leaky update, then a device-wide counter barrier.
// hidden[t] initially holds drive_t; it is overwritten with h_t.
// ---------------------------------------------------------------------------
__global__ void scan_kernel(const float* __restrict__ W_rec,
                            float*       __restrict__ hidden,
                            unsigned*    __restrict__ barrier,
                            int Tlen) {
    __shared__ float h_s[H];
    const int lane = threadIdx.x & 31;
    const int wv   = threadIdx.x >> 5;                  // 0..7
    const int rowbase = blockIdx.x * 32 + wv * 4;
    const unsigned nb = gridDim.x;

    for (int t = 0; t < Tlen; ++t) {
        // stage h_{t-1} (zeros at t==0) into LDS
        if (t == 0) {
            for (int i = threadIdx.x; i < H; i += blockDim.x) h_s[i] = 0.0f;
        } else {
            const float* hp = hidden + (size_t)(t - 1) * H;
            for (int i = threadIdx.x; i < H; i += blockDim.x) h_s[i] = hp[i];
        }
        __syncthreads();

        float hn[4];
#pragma unroll
        for (int rr = 0; rr < 4; ++rr) {
            const int r = rowbase + rr;
            const float4* wrow = (const float4*)(W_rec + (size_t)r * H);
            const float4* hv   = (const float4*)h_s;
            float4 acc = make_float4(0.f, 0.f, 0.f, 0.f);
#pragma unroll 4
            for (int i = 0; i < H / 128; ++i) {         // 16 iters, 512B/wave/iter
                const int idx = i * 32 + lane;
                const float4 w = wrow[idx];
                const float4 h = hv[idx];
                acc.x += w.x * h.x; acc.y += w.y * h.y;
                acc.z += w.z * h.z; acc.w += w.w * h.w;
            }
            float s = (acc.x + acc.y) + (acc.z + acc.w);
#pragma unroll
            for (int off = 16; off > 0; off >>= 1) s += __shfl_xor(s, off, 32);
            const float d = hidden[(size_t)t * H + r];  // drive value (not yet clobbered)
            const float x = s + d;
            const float I = (x > 30.0f) ? x : log1pf(expf(x));
            hn[rr] = (1.0f - ALPHA) * h_s[r] + ALPHA * I;
        }
        if (lane == 0)
            *(float4*)(hidden + (size_t)t * H + rowbase) =
                make_float4(hn[0], hn[1], hn[2], hn[3]);

        // device-wide barrier for step t
        __threadfence();
        __syncthreads();
        if (threadIdx.x == 0) {
            __hip_atomic_fetch_add(&barrier[t], 1u, __ATOMIC_ACQ_REL, __HIP_MEMORY_SCOPE_AGENT);
            while (__hip_atomic_load(&barrier[t], __ATOMIC_ACQUIRE, __HIP_MEMORY_SCOPE_AGENT) < nb)
                __builtin_amdgcn_s_sleep(1);
        }
        __syncthreads();
        __threadfence();                                // acquire before reading hidden[t]
    }
}

// ---------------------------------------------------------------------------
// Kernel 4: output = clip(hidden @ W_out^T + b_out). WMMA f32 16x16x4.
// One wave per 16(t) x 32(o) tile (both n-tiles, A loaded once), K double-
// buffered so WMMAs overlap the strided hidden-row loads.
// ---------------------------------------------------------------------------
__global__ void out_kernel(const float* __restrict__ hidden,
                           const float* __restrict__ W_out,
                           const float* __restrict__ b_out,
                           float*       __restrict__ out,
                           int Tlen) {
    const int lane = threadIdx.x & 31;
    const int t_tile = (blockIdx.x * blockDim.x + threadIdx.x) >> 5;  // wave id
    if (t_tile * 16 >= Tlen) return;

    const int m    = lane & 15;
    const int half = lane >> 4;
    int tt = t_tile * 16 + m;
    if (tt >= Tlen) tt = Tlen - 1;

    const float* arow = hidden + (size_t)tt * H;
    const float* br0  = W_out + (size_t)m        * H;   // o = m
    const float* br1  = W_out + (size_t)(m + 16) * H;   // o = m + 16

    v8f c0 = {}, c1 = {};

    int k = 2 * half;
    v2f a  = *(const v2f*)(arow + k);
    v2f b0 = *(const v2f*)(br0  + k);
    v2f b1 = *(const v2f*)(br1  + k);

#pragma unroll 8
    for (int k0 = 0; k0 < H; k0 += 4) {
        const int kn = ((k0 + 4) < H ? (k0 + 4) : 0) + 2 * half;
        v2f an  = *(const v2f*)(arow + kn);
        v2f b0n = *(const v2f*)(br0  + kn);
        v2f b1n = *(const v2f*)(br1  + kn);

        c0 = __builtin_amdgcn_wmma_f32_16x16x4_f32(false, a, false, b0, (short)0, c0, false, false);
        c1 = __builtin_amdgcn_wmma_f32_16x16x4_f32(false, a, false, b1, (short)0, c1, false, false);

        a = an; b0 = b0n; b1 = b1n;
    }

    const float bo0 = b_out[m];
    const float bo1 = b_out[m + 16];
#pragma unroll
    for (int i = 0; i < 8; ++i) {
        const int mo = i + 8 * half;
        const int to = t_tile * 16 + mo;
        if (to < Tlen) {
            float v0 = c0[i] + bo0;
            float v1 = c1[i] + bo1;
            v0 = fminf(fmaxf(v0, -1000.0f), 1000.0f);
            v1 = fminf(fmaxf(v1, -1000.0f), 1000.0f);
            out[(size_t)to * NOUT + m]      = v0;
            out[(size_t)to * NOUT + m + 16] = v1;
        }
    }
}

// ---------------------------------------------------------------------------
// d_in: 0=input_tensor(B,T,NIN) 1=ends 2=b 3=noise(T,H) 4=W_rec(H,H)
//       5=W_in(H,NIN) 6=b_h(H) 7=W_out(NOUT,H) 8=b_out(NOUT)
// d_out: [Tlen*NOUT output][Tlen*H hidden_all]
// ---------------------------------------------------------------------------
extern "C" void kernel_launch(void* const* d_in, const int* in_sizes, int n_in,
                              void* d_out, int out_size, void* d_ws, size_t ws_size,
                              hipStream_t stream) {
    const float* input_tensor = (const float*)d_in[0];
    const int*   bsel         = (const int*)  d_in[2];
    const float* noise        = (const float*)d_in[3];
    const float* W_rec        = (const float*)d_in[4];
    const float* W_in         = (const float*)d_in[5];
    const float* b_h          = (const float*)d_in[6];
    const float* W_out        = (const float*)d_in[7];
    const float* b_out        = (const float*)d_in[8];

    const int Tlen  = out_size / (NOUT + H);     // output shape encodes Tlen
    const int Tfull = in_sizes[3] / H;           // noise is (T, H)

    float* out    = (float*)d_out;
    float* hidden = out + (size_t)Tlen * NOUT;
    unsigned* barrier = (unsigned*)d_ws;         // Tlen counters (<= 32 KB)

    const int t_tiles = (Tlen + 15) / 16;

    // 1) drive -> hidden[]
    {
        const int waves  = t_tiles * (H / 64);
        const int blocks = (waves + 7) / 8;
        drive_kernel<<<blocks, 256, 0, stream>>>(input_tensor, bsel, noise, W_in,
                                                 b_h, hidden, Tlen, Tfull);
    }
    // 2) zero barrier counters
    zero_kernel<<<(Tlen + 255) / 256, 256, 0, stream>>>(barrier, Tlen);

    // 3) persistent sequential scan (64 blocks * 32 rows = H)
    scan_kernel<<<64, 256, 0, stream>>>(W_rec, hidden, barrier, Tlen);

    // 4) readout GEMM (one wave per 16x32 output tile)
    {
        const int waves  = t_tiles;
        const int blocks = (waves + 7) / 8;
        out_kernel<<<blocks, 256, 0, stream>>>(hidden, W_out, b_out, out, Tlen);
    }
}